// PartialVAE_38671885533517
// MI455X (gfx1250) — compile-verified
//
#include <hip/hip_runtime.h>

#define B_ 32
#define D_ 20000
#define K_ 16
#define HH_ 128
#define EH_ 256
#define L_ 32
#define DH_ 128

typedef __attribute__((ext_vector_type(2))) float v2f;
typedef __attribute__((ext_vector_type(8))) float v8f;

// ---------------- zero the c accumulator (B*16 floats) ----------------
__global__ void pvae_zero_c(float* __restrict__ c) {
    int t = threadIdx.x;
    if (t < B_ * K_) c[t] = 0.0f;
}

// ---------------- encoder: c[b,n] = sum_d mask * relu(relu(x*w1x + G[d]) @ W2 + b2) --------
// one wave (32 threads) per 16-row d-tile; f32 WMMA 16x16x4, K=128 (32 steps)
__global__ void __launch_bounds__(32) pvae_encoder(
    const float* __restrict__ x, const int* __restrict__ mask,
    const float* __restrict__ F, const float* __restrict__ fb,
    const float* __restrict__ W1, const float* __restrict__ b1,
    const float* __restrict__ W2, const float* __restrict__ b2,
    float* __restrict__ c_out)
{
    __shared__ float G[16][HH_];       // per-d part of layer-1 pre-activation
    __shared__ float w1x[HH_];         // W1 row 0 (x weight)
    __shared__ float W2s[HH_ * K_];    // 128x16
    __shared__ float b2s[K_];
    __shared__ float msk[16];
    __shared__ float red[32];

    const int lane = threadIdx.x;
    const int d0 = blockIdx.x * 16;

    // stage weights into LDS
    for (int i = lane; i < HH_; i += 32) w1x[i] = W1[i];
    if (lane < K_) b2s[lane] = b2[lane];
    for (int i = lane; i < HH_ * K_; i += 32) W2s[i] = W2[i];
    // build G for this 16-row tile: G[r][k] = b1[k] + fb*W1[17,k] + sum_j F[d,j]*W1[1+j,k]
    for (int i = lane; i < 16 * HH_; i += 32) {
        int r = i >> 7, k = i & 127;
        int d = d0 + r;
        float acc = b1[k] + fb[d] * W1[17 * HH_ + k];
        #pragma unroll
        for (int j = 0; j < K_; ++j) acc += F[d * K_ + j] * W1[(1 + j) * HH_ + k];
        G[r][k] = acc;
    }
    __syncthreads();

    const int row = lane & 15;
    const int khalf = (lane >> 4) << 1;   // 0 for lanes 0-15, 2 for lanes 16-31
    const int mbase = (lane >> 4) << 3;   // C rows handled by this half-wave

    for (int b = 0; b < B_; ++b) {
        float xv = x[b * D_ + d0 + row];
        if (lane < 16) msk[lane] = (float)mask[b * D_ + d0 + lane];
        __syncthreads();

        v8f acc = {};
        #pragma unroll
        for (int s = 0; s < 32; ++s) {
            int k = 4 * s + khalf;
            // A fragment: h1[row][k], h1[row][k+1]  (rank-1 layer-1 + relu)
            float a0 = fmaxf(fmaf(xv, w1x[k],     G[row][k]),     0.0f);
            float a1 = fmaxf(fmaf(xv, w1x[k + 1], G[row][k + 1]), 0.0f);
            v2f A = {a0, a1};
            // B fragment: W2[k][N], W2[k+1][N], N = lane&15
            v2f Bf = {W2s[k * K_ + row], W2s[(k + 1) * K_ + row]};
            acc = __builtin_amdgcn_wmma_f32_16x16x4_f32(
                false, A, false, Bf, (short)0, acc, false, false);
        }

        // epilogue: relu(+b2), mask, reduce the 16 rows into c[b][N]
        float sum = 0.0f;
        #pragma unroll
        for (int r = 0; r < 8; ++r) {
            float v = fmaxf(acc[r] + b2s[row], 0.0f);
            sum += v * msk[mbase + r];
        }
        red[lane] = sum;
        __syncthreads();
        if (lane < 16) atomicAdd(&c_out[b * K_ + lane], red[lane] + red[lane + 16]);
        __syncthreads();
    }
}

// ---------------- Ht[n][d] = F[d]@Wj1[128:144,n] + fb[d]*Wj1[144,n] + bj1[n] -------------
__global__ void pvae_ht(const float* __restrict__ F, const float* __restrict__ fb,
                        const float* __restrict__ Wj1, const float* __restrict__ bj1,
                        float* __restrict__ Ht)
{
    int idx = blockIdx.x * blockDim.x + threadIdx.x;   // n-major for coalesced writes
    if (idx >= DH_ * D_) return;
    int n = idx / D_;
    int d = idx - n * D_;
    float acc = bj1[n] + fb[d] * Wj1[(HH_ + K_) * DH_ + n];
    #pragma unroll
    for (int j = 0; j < K_; ++j) acc += F[d * K_ + j] * Wj1[(HH_ + j) * DH_ + n];
    Ht[n * D_ + d] = acc;
}

// ---------------- bottleneck: c -> enc -> (mu,logvar) -> pz -> P ----------------
__global__ void __launch_bounds__(256) pvae_mid(
    const float* __restrict__ c_ws,
    const float* __restrict__ We1, const float* __restrict__ be1,
    const float* __restrict__ We2, const float* __restrict__ be2,
    const float* __restrict__ Wz,  const float* __restrict__ bz,
    const float* __restrict__ Wj1, float* __restrict__ P_ws,
    float* __restrict__ out)
{
    __shared__ float cL[K_];
    __shared__ float s1[EH_];
    __shared__ float muL[L_];
    __shared__ float pzL[DH_];
    const int t = threadIdx.x;
    const int REC = B_ * D_;

    for (int b = 0; b < B_; ++b) {
        if (t < K_) cL[t] = c_ws[b * K_ + t];
        __syncthreads();
        // layer e1: 16 -> 256
        float a = be1[t];
        #pragma unroll
        for (int j = 0; j < K_; ++j) a += cL[j] * We1[j * EH_ + t];
        s1[t] = fmaxf(a, 0.0f);
        __syncthreads();
        // layer e2: 256 -> 64 ; split mu / logvar
        if (t < 2 * L_) {
            float e = be2[t];
            for (int j = 0; j < EH_; ++j) e += s1[j] * We2[j * (2 * L_) + t];
            if (t < L_) { out[REC + b * L_ + t] = e; muL[t] = e; }
            else        { out[REC + B_ * L_ + b * L_ + (t - L_)] = e; }
        }
        __syncthreads();
        // pz = relu(mu @ Wz + bz): 32 -> 128
        if (t < DH_) {
            float p = bz[t];
            #pragma unroll
            for (int j = 0; j < L_; ++j) p += muL[j] * Wz[j * DH_ + t];
            pzL[t] = fmaxf(p, 0.0f);
        }
        __syncthreads();
        // P[b] = pz @ Wj1[0:128,:]
        if (t < DH_) {
            float q = 0.0f;
            for (int j = 0; j < DH_; ++j) q += pzL[j] * Wj1[j * DH_ + t];
            P_ws[b * DH_ + t] = q;
        }
        __syncthreads();
    }
}

// ---------------- decoder: rec[b,d] = relu(P[b]+Ht[:,d]) . Wj2 + bj2 ----------------
__global__ void __launch_bounds__(256) pvae_decoder(
    const float* __restrict__ P, const float* __restrict__ Ht,
    const float* __restrict__ Wj2, const float* __restrict__ bj2,
    float* __restrict__ rec)
{
    __shared__ float w2s[DH_];
    int t = threadIdx.x;
    if (t < DH_) w2s[t] = Wj2[t];
    __syncthreads();

    int idx = blockIdx.x * blockDim.x + t;
    if (idx >= B_ * D_) return;
    int b = idx / D_;
    int d = idx - b * D_;
    const float* Pb = P + b * DH_;
    float acc = bj2[0];
    #pragma unroll 8
    for (int n = 0; n < DH_; ++n)
        acc += fmaxf(Pb[n] + Ht[n * D_ + d], 0.0f) * w2s[n];
    rec[idx] = acc;
}

extern "C" void kernel_launch(void* const* d_in, const int* in_sizes, int n_in,
                              void* d_out, int out_size, void* d_ws, size_t ws_size,
                              hipStream_t stream) {
    const float* x   = (const float*)d_in[0];
    const int*   msk = (const int*)  d_in[1];
    const float* F   = (const float*)d_in[2];
    const float* fb  = (const float*)d_in[3];
    const float* W1  = (const float*)d_in[4];
    const float* b1  = (const float*)d_in[5];
    const float* W2  = (const float*)d_in[6];
    const float* b2  = (const float*)d_in[7];
    const float* We1 = (const float*)d_in[8];
    const float* be1 = (const float*)d_in[9];
    const float* We2 = (const float*)d_in[10];
    const float* be2 = (const float*)d_in[11];
    const float* Wz  = (const float*)d_in[12];
    const float* bz  = (const float*)d_in[13];
    const float* Wj1 = (const float*)d_in[14];
    const float* bj1 = (const float*)d_in[15];
    const float* Wj2 = (const float*)d_in[16];
    const float* bj2 = (const float*)d_in[17];

    float* out  = (float*)d_out;
    float* ws   = (float*)d_ws;
    float* c_ws = ws;                       // B*16 = 512 floats
    float* P_ws = ws + 512;                 // B*128 = 4096 floats
    float* Ht   = ws + 512 + 4096;          // 128*D = 2,560,000 floats (~10.2 MB)

    pvae_zero_c<<<1, 512, 0, stream>>>(c_ws);
    pvae_encoder<<<D_ / 16, 32, 0, stream>>>(x, msk, F, fb, W1, b1, W2, b2, c_ws);
    pvae_ht<<<(DH_ * D_) / 256, 256, 0, stream>>>(F, fb, Wj1, bj1, Ht);
    pvae_mid<<<1, 256, 0, stream>>>(c_ws, We1, be1, We2, be2, Wz, bz, Wj1, P_ws, out);
    pvae_decoder<<<(B_ * D_ + 255) / 256, 256, 0, stream>>>(P_ws, Ht, Wj2, bj2, out);
}